// BilinearInterpolation_60670708023631
// MI455X (gfx1250) — compile-verified
//
#include <hip/hip_runtime.h>

// Problem constants (from the reference).
namespace {
constexpr int kOutHW = 64;           // OUT_H == OUT_W == 64
constexpr int kDepth = 65;
constexpr int kImgH  = 128;
constexpr int kImgW  = 128;
constexpr int kC     = 32;           // channels == wave32 width
constexpr int kPadZ  = (kDepth - 1) / 2;  // image slab lives at z == 32
}  // namespace

__device__ __forceinline__ int clampi(int v, int lo, int hi) {
  return v < lo ? lo : (v > hi ? hi : v);
}

#if __has_builtin(__builtin_amdgcn_global_load_async_to_lds_b32)
#define WARP_ASYNC_LDS 1
#else
#define WARP_ASYNC_LDS 0
#endif

// One wave (32 lanes) per output pixel; lane == channel. 8 waves per block.
__global__ __launch_bounds__(256) void warp_gather_kernel(
    const float* __restrict__ img,    // (128,128,32) fp32
    const float* __restrict__ trans,  // (12,) fp32
    float* __restrict__ out) {        // (64,64,32) fp32
  __shared__ float sT[16];

  // --- Stage the 3x4 transform through LDS (CDNA5 async path if available).
#if WARP_ASYNC_LDS
  if (threadIdx.x < 12u) {
    __builtin_amdgcn_global_load_async_to_lds_b32(
        (int*)(trans + threadIdx.x), (int*)&sT[threadIdx.x], 0, 0);
  }
#if __has_builtin(__builtin_amdgcn_s_wait_asynccnt)
  __builtin_amdgcn_s_wait_asynccnt(0);
#else
  asm volatile("s_wait_asynccnt 0x0" ::: "memory");
#endif
#else
  if (threadIdx.x < 12u) sT[threadIdx.x] = trans[threadIdx.x];
#endif
  __syncthreads();

  const int lane = (int)(threadIdx.x & 31u);
  const int pix  = (int)blockIdx.x * 8 + (int)(threadIdx.x >> 5);  // 0..4095
  const int oy   = pix >> 6;
  const int ox   = pix & (kOutHW - 1);

  // linspace(-1, 1, 64)[j] == j * (2/63) - 1
  const float gx = (float)ox * (2.0f / 63.0f) - 1.0f;
  const float gy = (float)oy * (2.0f / 63.0f) - 1.0f;

  const float T00 = sT[0], T01 = sT[1], T02 = sT[2],  T03 = sT[3];
  const float T10 = sT[4], T11 = sT[5], T12 = sT[6],  T13 = sT[7];
  const float T20 = sT[8], T21 = sT[9], T22 = sT[10], T23 = sT[11];

  // Parts of T*[gx,gy,gz,1] that do not depend on gz.
  const float xb = T00 * gx + T01 * gy + T03;
  const float yb = T10 * gx + T11 * gy + T13;
  const float zb = T20 * gx + T21 * gy + T23;

  float acc = 0.0f;

#pragma unroll 1
  for (int k = 0; k < kDepth; ++k) {
    const float gz = (float)k * (2.0f / 64.0f) - 1.0f;  // linspace(-1,1,65)
    // x = 0.5*(sx+1)*128 = 64*sx + 64 ; z = 0.5*(sz+1)*65 = 32.5*sz + 32.5
    const float x = 64.0f * (xb + T02 * gz) + 64.0f;
    const float y = 64.0f * (yb + T12 * gz) + 64.0f;
    const float z = 32.5f * (zb + T22 * gz) + 32.5f;

    // astype(int32) truncates toward zero, then clip -- match exactly.
    const int zi = (int)z;
    const int z0 = clampi(zi, 0, kDepth - 1);
    const int z1 = clampi(zi + 1, 0, kDepth - 1);

    // Padded volume is zero except z == kPadZ: fold the z-sum into wz.
    float wz = 0.0f;
    if (z0 == kPadZ) wz += (float)z1 - z;
    if (z1 == kPadZ) wz += z - (float)z0;

    if (wz != 0.0f) {  // wave-uniform skip: ~95% of depth samples contribute 0
      const int xi = (int)x;
      const int yi = (int)y;
      const int x0 = clampi(xi, 0, kImgW - 1);
      const int x1 = clampi(xi + 1, 0, kImgW - 1);
      const int y0 = clampi(yi, 0, kImgH - 1);
      const int y1 = clampi(yi + 1, 0, kImgH - 1);

      // Weights use clipped float coords (faithful to the reference), and the
      // reference pairs fx with the y-index and fy with the x-index.
      const float fx0 = (float)x1 - x;
      const float fx1 = x - (float)x0;
      const float fy0 = (float)y1 - y;
      const float fy1 = y - (float)y0;

      const float* r00 = img + (size_t)((y0 * kImgW + x0) * kC);
      const float* r01 = img + (size_t)((y0 * kImgW + x1) * kC);
      const float* r10 = img + (size_t)((y1 * kImgW + x0) * kC);
      const float* r11 = img + (size_t)((y1 * kImgW + x1) * kC);

      // lane == channel: each load is one fully-coalesced 128B cacheline.
      const float v00 = r00[lane];
      const float v01 = r01[lane];
      const float v10 = r10[lane];
      const float v11 = r11[lane];

      acc += wz * (fx0 * fy0 * v00 + fx0 * fy1 * v01 +
                   fx1 * fy0 * v10 + fx1 * fy1 * v11);
    }
  }

  out[(size_t)pix * kC + lane] = acc;  // coalesced 128B store per wave
}

extern "C" void kernel_launch(void* const* d_in, const int* in_sizes, int n_in,
                              void* d_out, int out_size, void* d_ws, size_t ws_size,
                              hipStream_t stream) {
  (void)in_sizes; (void)n_in; (void)out_size; (void)d_ws; (void)ws_size;
  const float* img   = (const float*)d_in[0];  // (1,128,128,32) fp32
  const float* trans = (const float*)d_in[1];  // (1,12) fp32
  float* out = (float*)d_out;                  // (1,64,64,32) fp32

  // 4096 output pixels, 8 per block (8 waves of 32 lanes).
  dim3 grid(512), block(256);
  hipLaunchKernelGGL(warp_gather_kernel, grid, block, 0, stream, img, trans, out);
}